// SimpleDICK_6390911336717
// MI455X (gfx1250) — compile-verified
//
#include <hip/hip_runtime.h>
#include <stdint.h>

// ---------------------------------------------------------------------------
// SimpleDICK: separable 3x3 conv (1x3 then 3x1, zero pad) + tridiagonal
// log-det recurrence.  Memory-bound (1.5 FLOP/B vs ~3.4 FLOP/B machine
// balance at 23.3 TB/s) -> optimize the data path:
//   * interior tiles: one TDM TENSOR_LOAD_TO_LDS descriptor per tile
//     (DMA engine does all address generation; right/bottom halo zero-filled
//     by TDM OOB semantics), drained with s_wait_tensorcnt.
//   * top/left edge tiles: gfx1250 async global->LDS b128 staging.
//   * compute: two-pass separable stencil in LDS, b128 ds loads, b128 stores.
// ---------------------------------------------------------------------------

#define IMG_H   1024
#define IMG_W   1024
#define N_IMG   16
#define HW      (IMG_H * IMG_W)
#define TILE    64          // 64x64 outputs per workgroup
#define HALO_H  66          // TILE + 2 rows
#define XS_W    72          // staged cols: global c0-4 .. c0+67 (18 b128 blocks)
#define XBLK    18          // b128 blocks per staged row
#define YS_W    64

typedef uint32_t v4u __attribute__((ext_vector_type(4)));
typedef int      v8i __attribute__((ext_vector_type(8)));
typedef int      v4i __attribute__((ext_vector_type(4)));

__global__ __launch_bounds__(256) void dick_conv_kernel(
    const float* __restrict__ x,
    const float* __restrict__ hk,
    const float* __restrict__ vk,
    float* __restrict__ out) {
  // xs[row][col] = x[r0-1+row][c0-4+col]   (zero where OOB)
  __shared__ __align__(16) float xs[HALO_H * XS_W];   // 19.0 KB (packed TDM tile)
  __shared__ __align__(16) float ys[HALO_H * YS_W];   // 16.9 KB

  const int tid = threadIdx.x;
  const int b   = blockIdx.x;            // 16 imgs * 16 * 16 tiles = 4096
  const int img = b >> 8;
  const int tr  = (b >> 4) & 15;
  const int tc  = b & 15;
  const int r0  = tr * TILE;
  const int c0  = tc * TILE;

  const float h0 = hk[0], h1 = hk[1], h2 = hk[2];
  const float v0 = vk[0], v1 = vk[1], v2 = vk[2];

  const float*   xbase = x + (size_t)img * HW;
  const uint64_t sbase = (uint64_t)(uintptr_t)xbase;

  // ---- Phase 1: stage 66x72 halo tile into LDS --------------------------
  if (tr > 0 && tc > 0) {
    // TDM fast path: staged origin (r0-1, c0-4) is in-bounds; right/bottom
    // overhang (tc==15 / tr==15) is zero-filled by TDM OOB reads.
    if (tid < 32) {                      // one wave builds D# and issues DMA
      const uint64_t gaddr =
          (uint64_t)(uintptr_t)(xbase + (size_t)(r0 - 1) * IMG_W + (c0 - 4));
      const uint32_t ldsa = (uint32_t)(uintptr_t)&xs[0];
      const uint32_t td0  = (uint32_t)(IMG_W - (c0 - 4));  // cols to image edge
      const uint32_t td1  = (uint32_t)(IMG_H - (r0 - 1));  // rows to image edge
      v4u g0;
      g0[0] = 1u;                                    // count=1, user desc
      g0[1] = ldsa;                                  // lds_addr [63:32]
      g0[2] = (uint32_t)gaddr;                       // global_addr[31:0]
      g0[3] = (uint32_t)((gaddr >> 32) & 0x01ffffffu) | 0x80000000u; // +type=2
      v8i g1;
      g1[0] = (int)(2u << 16);                       // data_size=4B, mask=0
      g1[1] = (int)((td0 & 0xffffu) << 16);          // tensor_dim0[15:0]
      g1[2] = (int)((td0 >> 16) | ((td1 & 0xffffu) << 16));  // d0 hi | d1 lo
      g1[3] = (int)((td1 >> 16) | ((uint32_t)XS_W << 16));   // d1 hi | tile_dim0
      g1[4] = (int)HALO_H;                           // tile_dim1; tile_dim2=0
      g1[5] = (int)IMG_W;                            // tensor_dim0_stride lo
      g1[6] = 0;                                     // stride hi | dim1_stride
      g1[7] = 0;
      const v4i gz  = {0, 0, 0, 0};                  // groups 2/3 unused (2D)
      const v8i gz8 = {0, 0, 0, 0, 0, 0, 0, 0};      // extra group (clang-23 form)
      __builtin_amdgcn_tensor_load_to_lds(g0, g1, gz, gz, gz8, 0);
      __builtin_amdgcn_s_wait_tensorcnt(0);
    }
  } else {
    // Async fallback for tiles whose halo starts before the image.
    // Every b128 granule is fully in-bounds or fully OOB (c0%64==0, W%4==0).
    for (int u = tid; u < HALO_H * XBLK; u += 256) {
      const int row = u / XBLK;
      const int k   = u - row * XBLK;
      const int gr  = r0 - 1 + row;
      const int gc4 = c0 - 4 + (k << 2);
      float* ldsp = &xs[row * XS_W + (k << 2)];
      if ((unsigned)gr < IMG_H && (unsigned)gc4 < IMG_W) {
        const uint32_t lds_addr = (uint32_t)(uintptr_t)ldsp;
        const uint32_t voff     = (uint32_t)((gr * IMG_W + gc4) * 4);
        asm volatile("global_load_async_to_lds_b128 %0, %1, %2 offset:0"
                     :
                     : "v"(lds_addr), "v"(voff), "s"(sbase)
                     : "memory");
      } else {
        *(float4*)ldsp = make_float4(0.f, 0.f, 0.f, 0.f);  // zero padding
      }
    }
    asm volatile("s_wait_asynccnt 0" ::: "memory");
  }
  __syncthreads();

  // ---- Phase 2: horizontal 1x3 pass, float4-vectorized, into ys -----------
  // y[cc] = h0*xs[cc+3] + h1*xs[cc+4] + h2*xs[cc+5]  (cc = output col 0..63)
  for (int idx = tid; idx < HALO_H * (TILE / 4); idx += 256) {
    const int row = idx >> 4;            // 0..65
    const int cg  = (idx & 15) << 2;     // 0,4,...,60
    const float* xrow = &xs[row * XS_W];
    const float  aw = xrow[cg + 3];                    // ds_load_b32
    const float4 B  = *(const float4*)&xrow[cg + 4];   // ds_load_b128 (aligned)
    const float  cx = xrow[cg + 8];                    // ds_load_b32
    float4 y;
    y.x = fmaf(h2, B.y, fmaf(h1, B.x, h0 * aw));
    y.y = fmaf(h2, B.z, fmaf(h1, B.y, h0 * B.x));
    y.z = fmaf(h2, B.w, fmaf(h1, B.z, h0 * B.y));
    y.w = fmaf(h2, cx,  fmaf(h1, B.w, h0 * B.z));
    *(float4*)&ys[row * YS_W + cg] = y;
  }
  __syncthreads();

  // ---- Phase 3: vertical 3x1 pass, float4 ds_load_b128 + global b128 ------
  // Thread t: cols cg..cg+3, rows rg..rg+3 (reads ys rows rg..rg+5).
  const int cg = (tid & 15) << 2;        // 0..60
  const int rg = (tid >> 4) << 2;        // 0..60
  float4 Y[6];
#pragma unroll
  for (int k = 0; k < 6; ++k)
    Y[k] = *(const float4*)&ys[(rg + k) * YS_W + cg];

  float* obase = out + (size_t)img * HW;
#pragma unroll
  for (int r = 0; r < 4; ++r) {
    const float4 a = Y[r], bb = Y[r + 1], c = Y[r + 2];
    float4 z;
    z.x = fmaf(v2, c.x, fmaf(v1, bb.x, v0 * a.x));
    z.y = fmaf(v2, c.y, fmaf(v1, bb.y, v0 * a.y));
    z.z = fmaf(v2, c.z, fmaf(v1, bb.z, v0 * a.z));
    z.w = fmaf(v2, c.w, fmaf(v1, bb.w, v0 * a.w));
    *(float4*)&obase[(size_t)(r0 + rg + r) * IMG_W + c0 + cg] = z;
  }
}

// ---- log-det: two length-1022 continuant recurrences, one lane ------------
__global__ void dick_logdet_kernel(const float* __restrict__ hk,
                                   const float* __restrict__ vk,
                                   float* __restrict__ out) {
  if (threadIdx.x != 0 || blockIdx.x != 0) return;
  float total = 0.0f;
#pragma unroll 1
  for (int which = 0; which < 2; ++which) {
    const float* k = (which == 0) ? vk : hk;   // order irrelevant (sum)
    const float k0 = k[0], k1 = k[1], k2 = k[2];
    const float mx = fmaxf(fabsf(k0), fmaxf(fabsf(k1), fabsf(k2)));
    const float bn = k0 / mx, a = k1 / mx, cn = k2 / mx;
    const float bc = bn * cn;
    float d0 = 1.0f, d1 = a;
#pragma unroll 1
    for (int i = 0; i < IMG_H - 2; ++i) {      // size-2 = 1022 steps
      const float nd = fmaf(a, d1, -bc * d0);
      d0 = d1;
      d1 = nd;
    }
    // blocks * (size*log(mx) + log(|det|+1e-8)); size = blocks = 1024
    total += 1024.0f * (1024.0f * logf(mx) + logf(fabsf(d1) + 1e-8f));
  }
  out[(size_t)N_IMG * HW] = total;   // element 16777216
}

extern "C" void kernel_launch(void* const* d_in, const int* in_sizes, int n_in,
                              void* d_out, int out_size, void* d_ws,
                              size_t ws_size, hipStream_t stream) {
  (void)in_sizes; (void)n_in; (void)out_size; (void)d_ws; (void)ws_size;
  const float* x  = (const float*)d_in[0];
  const float* hk = (const float*)d_in[1];
  const float* vk = (const float*)d_in[2];
  float* out = (float*)d_out;

  dick_conv_kernel<<<dim3(N_IMG * 16 * 16), dim3(256), 0, stream>>>(x, hk, vk, out);
  dick_logdet_kernel<<<dim3(1), dim3(32), 0, stream>>>(hk, vk, out);
}